// GraphAttention_88502096101456
// MI455X (gfx1250) — compile-verified
//
#include <hip/hip_runtime.h>

#define IN_F  256
#define OUT_F 128
#define LRELU_ALPHA 0.2f

typedef __attribute__((ext_vector_type(2))) float v2f;
typedef __attribute__((ext_vector_type(8))) float v8f;

// ---------- atomic helpers (relaxed/agent -> native global_atomic_* ops) ----------
__device__ __forceinline__ void atomAddF32(float* p, float v) {
  __hip_atomic_fetch_add(p, v, __ATOMIC_RELAXED, __HIP_MEMORY_SCOPE_AGENT);
}
__device__ __forceinline__ void atomMaxU32(unsigned* p, unsigned v) {
  __hip_atomic_fetch_max(p, v, __ATOMIC_RELAXED, __HIP_MEMORY_SCOPE_AGENT);
}
// order-preserving float<->uint mapping for atomic float max
__device__ __forceinline__ unsigned encOrd(float f) {
  unsigned u = __float_as_uint(f);
  return (u & 0x80000000u) ? ~u : (u | 0x80000000u);
}
__device__ __forceinline__ float decOrd(unsigned u) {
  return (u & 0x80000000u) ? __uint_as_float(u & 0x7FFFFFFFu)
                           : __uint_as_float(~u);
}

// ---------- kernel 0: zero-init accumulators ----------
__global__ void k_init(float* __restrict__ out, float* __restrict__ e_sum,
                       unsigned* __restrict__ emax, int N) {
  size_t i = (size_t)blockIdx.x * blockDim.x + threadIdx.x;
  size_t tot = (size_t)N * OUT_F;
  if (i < tot) out[i] = 0.0f;
  if (i < (size_t)N) { e_sum[i] = 0.0f; emax[i] = 0u; }  // 0 < encOrd(-inf)
}

// ---------- kernel 1: Wh = x @ W via V_WMMA_F32_16X16X4_F32 ----------
// block = 256 threads = 8 waves; block b owns rows [16b,16b+16); wave w owns cols [16w,16w+16)
__global__ void k_gemm_wmma(const float* __restrict__ x, const float* __restrict__ W,
                            float* __restrict__ Wh) {
  const int lane = threadIdx.x & 31;
  const int wave = threadIdx.x >> 5;        // 0..7 -> column tile
  const int row0 = blockIdx.x * 16;
  const int col0 = wave * 16;
  const int r  = lane & 15;                 // row/col within tile
  const int kh = lane >> 4;                 // 0: K={0,1}, 1: K={2,3}

  // A fragment source: x[row0+r, k + 2*kh + {0,1}]  (8B-aligned pair)
  const float* xrow = x + (size_t)(row0 + r) * IN_F + 2 * kh;
  // B fragment source: W[k + 2*kh + {0,1}, col0 + r]
  const float* wcol = W + (size_t)(2 * kh) * OUT_F + col0 + r;

  v8f c = {};
#pragma unroll
  for (int k = 0; k < IN_F; k += 4) {
    const float2 ap = *(const float2*)(xrow + k);
    v2f a; a.x = ap.x; a.y = ap.y;
    v2f b;
    b.x = wcol[(size_t)k * OUT_F];
    b.y = wcol[(size_t)k * OUT_F + OUT_F];
    c = __builtin_amdgcn_wmma_f32_16x16x4_f32(
        /*neg_a=*/false, a, /*neg_b=*/false, b,
        /*c_mod=*/(short)0, c, /*reuse_a=*/false, /*reuse_b=*/false);
  }
  // D layout: VGPR v -> row = v + 8*kh, col = r
  float* o = Wh + (size_t)(row0 + 8 * kh) * OUT_F + col0 + r;
#pragma unroll
  for (int v = 0; v < 8; ++v) o[(size_t)v * OUT_F] = c[v];
}

// ---------- kernel 2: e_left / e_right per-node dots (one wave per node) ----------
__global__ void k_dots(const float* __restrict__ Wh, const float* __restrict__ a_l,
                       const float* __restrict__ a_r, float* __restrict__ e_l,
                       float* __restrict__ e_r, int N) {
  const int gw   = (int)(((size_t)blockIdx.x * blockDim.x + threadIdx.x) >> 5);
  const int lane = threadIdx.x & 31;
  if (gw >= N) return;
  const float* row = Wh + (size_t)gw * OUT_F;
  float sl = 0.0f, sr = 0.0f;
#pragma unroll
  for (int i = 0; i < OUT_F / 32; ++i) {
    const int j = lane + 32 * i;
    const float w = row[j];
    sl = fmaf(w, a_l[j], sl);
    sr = fmaf(w, a_r[j], sr);
  }
#pragma unroll
  for (int off = 16; off > 0; off >>= 1) {
    sl += __shfl_xor(sl, off, 32);
    sr += __shfl_xor(sr, off, 32);
  }
  if (lane == 0) { e_l[gw] = sl; e_r[gw] = sr; }
}

// ---------- kernel 3: edge logits + segment max ----------
__global__ void k_edge_max(const int* __restrict__ src, const int* __restrict__ dst,
                           const float* __restrict__ e_l, const float* __restrict__ e_r,
                           unsigned* __restrict__ emax, float* __restrict__ e_store, int E) {
  const int e = blockIdx.x * blockDim.x + threadIdx.x;
  if (e >= E) return;
  const int s = src[e], d = dst[e];
  float v = e_l[s] + e_r[d];
  v = v > 0.0f ? v : LRELU_ALPHA * v;       // LeakyReLU
  e_store[e] = v;
  atomMaxU32(&emax[s], encOrd(v));
}

// ---------- kernel 4: exp + segment sum ----------
__global__ void k_edge_sum(const int* __restrict__ src, const float* __restrict__ e_store,
                           const unsigned* __restrict__ emax, float* __restrict__ e_exp,
                           float* __restrict__ e_sum, int E) {
  const int e = blockIdx.x * blockDim.x + threadIdx.x;
  if (e >= E) return;
  const int s = src[e];
  const float m  = decOrd(emax[s]);
  const float ex = __expf(e_store[e] - m);
  e_exp[e] = ex;
  atomAddF32(&e_sum[s], ex);
}

// ---------- kernel 5: weighted scatter (one wave per edge, float4 per lane) ----------
__global__ void k_scatter(const int* __restrict__ src, const int* __restrict__ dst,
                          const float* __restrict__ e_exp, const float* __restrict__ e_sum,
                          const float* __restrict__ Wh, float* __restrict__ out, int E) {
  const long long gw = (((long long)blockIdx.x * blockDim.x) + threadIdx.x) >> 5;
  const int lane = threadIdx.x & 31;
  if (gw >= E) return;
  const int s = src[gw], d = dst[gw];
  const float alpha = e_exp[gw] / (e_sum[s] + 1e-9f);
  const float4 m = ((const float4*)(Wh + (size_t)d * OUT_F))[lane];
  float* orow = out + (size_t)s * OUT_F + 4 * lane;
  atomAddF32(&orow[0], alpha * m.x);
  atomAddF32(&orow[1], alpha * m.y);
  atomAddF32(&orow[2], alpha * m.z);
  atomAddF32(&orow[3], alpha * m.w);
}

// ---------- kernel 6: final ReLU ----------
__global__ void k_relu(float* __restrict__ out, size_t n) {
  const size_t i = (size_t)blockIdx.x * blockDim.x + threadIdx.x;
  if (i < n) out[i] = fmaxf(out[i], 0.0f);
}

extern "C" void kernel_launch(void* const* d_in, const int* in_sizes, int n_in,
                              void* d_out, int out_size, void* d_ws, size_t ws_size,
                              hipStream_t stream) {
  const float* x   = (const float*)d_in[0];
  const int*   ei  = (const int*)d_in[1];
  const float* W   = (const float*)d_in[2];
  const float* a_l = (const float*)d_in[3];
  const float* a_r = (const float*)d_in[4];
  float* out = (float*)d_out;

  const int N = in_sizes[0] / IN_F;   // 100000
  const int E = in_sizes[1] / 2;      // 1600000
  const int* src = ei;                // edge_index[0]
  const int* dst = ei + E;            // edge_index[1]

  // workspace layout
  char* ws = (char*)d_ws;
  float*    Wh      = (float*)ws;     ws += (size_t)N * OUT_F * sizeof(float);
  float*    e_l     = (float*)ws;     ws += (size_t)N * sizeof(float);
  float*    e_r     = (float*)ws;     ws += (size_t)N * sizeof(float);
  unsigned* emax    = (unsigned*)ws;  ws += (size_t)N * sizeof(unsigned);
  float*    e_sum   = (float*)ws;     ws += (size_t)N * sizeof(float);
  float*    e_store = (float*)ws;     ws += (size_t)E * sizeof(float);
  float*    e_exp   = (float*)ws;     ws += (size_t)E * sizeof(float);

  const int BLK = 256;

  // 0) zero out accumulators
  {
    size_t tot = (size_t)N * OUT_F;
    k_init<<<(unsigned)((tot + BLK - 1) / BLK), BLK, 0, stream>>>(out, e_sum, emax, N);
  }
  // 1) Wh = x @ W   (N divisible by 16: 100000 = 6250*16)
  k_gemm_wmma<<<N / 16, BLK, 0, stream>>>(x, W, Wh);
  // 2) per-node attention dots
  {
    size_t thr = (size_t)N * 32;
    k_dots<<<(unsigned)((thr + BLK - 1) / BLK), BLK, 0, stream>>>(Wh, a_l, a_r, e_l, e_r, N);
  }
  // 3) edge logits + segment max
  k_edge_max<<<(E + BLK - 1) / BLK, BLK, 0, stream>>>(src, dst, e_l, e_r, emax, e_store, E);
  // 4) exp + segment sum
  k_edge_sum<<<(E + BLK - 1) / BLK, BLK, 0, stream>>>(src, e_store, emax, e_exp, e_sum, E);
  // 5) weighted message scatter
  {
    size_t thr = (size_t)E * 32;
    k_scatter<<<(unsigned)((thr + BLK - 1) / BLK), BLK, 0, stream>>>(src, dst, e_exp, e_sum, Wh, out, E);
  }
  // 6) final ReLU
  {
    size_t tot = (size_t)N * OUT_F;
    k_relu<<<(unsigned)((tot + BLK - 1) / BLK), BLK, 0, stream>>>(out, tot);
  }
}